// SurveyEmbeddings_24988119728796
// MI455X (gfx1250) — compile-verified
//
#include <hip/hip_runtime.h>
#include <hip/hip_bf16.h>
#include <stdint.h>

// ---------------------------------------------------------------------------
// SurveyEmbeddings forward for MI455X (gfx1250).
// Memory-bound kernel (~272 MB HBM traffic, ~12 us floor at 23.3 TB/s).
// No matmul exists in this op -> WMMA not applicable; the CDNA5-specific path
// used here is the Tensor Data Mover (tensor_load_to_lds + s_wait_tensorcnt)
// to stage the reused q_emb tile into LDS, plus B128 vector memory ops and
// nontemporal streaming stores for the 256 MB output.
// ---------------------------------------------------------------------------

#define B_TOT   4096
#define Q_TOT   1024
#define D_DIM   16
#define BT      16     // batch rows per block
#define QT      256    // questions per block
#define THREADS 256
#define LN_EPS  1e-5f

typedef float        f32x4 __attribute__((ext_vector_type(4)));
typedef unsigned int u32x4 __attribute__((ext_vector_type(4)));
typedef int          i32x4 __attribute__((ext_vector_type(4)));
typedef int          i32x8 __attribute__((ext_vector_type(8)));

#if defined(__HIP_DEVICE_COMPILE__) && __has_builtin(__builtin_amdgcn_tensor_load_to_lds)
#define USE_TDM 1
#else
#define USE_TDM 0
#endif

__device__ __forceinline__ void wait_tensorcnt0() {
#if defined(__HIP_DEVICE_COMPILE__) && __has_builtin(__builtin_amdgcn_s_wait_tensorcnt)
    __builtin_amdgcn_s_wait_tensorcnt(0);
#elif defined(__HIP_DEVICE_COMPILE__)
    asm volatile("s_wait_tensorcnt 0x0" ::: "memory");
#endif
}

// Issue one TDM descriptor: contiguous 1-D copy of `nelem` f32 from global
// `gsrc` into LDS byte offset `lds_off`. D# packing per CDNA5 ISA ch.8.
__device__ __forceinline__ void tdm_load_1d_f32(const float* gsrc,
                                                unsigned lds_off,
                                                unsigned nelem) {
#if USE_TDM
    const uint64_t ga = (uint64_t)(uintptr_t)gsrc;
    u32x4 g0;
    g0.x = 1u;                                        // count=1 (valid user D#)
    g0.y = lds_off;                                   // lds_addr (bytes)
    g0.z = (unsigned)(ga & 0xFFFFFFFFull);            // global_addr[31:0]
    g0.w = (unsigned)((ga >> 32) & 0x01FFFFFFull)     // global_addr[56:32]
         | (2u << 30);                                // type = 2 ("image")
    i32x8 g1;
    g1[0] = (int)(2u << 16);                          // data_size = 2 (4 bytes)
    g1[1] = (int)((nelem & 0xFFFFu) << 16);           // tensor_dim0[15:0]
    g1[2] = (int)(((nelem >> 16) & 0xFFFFu)           // tensor_dim0[31:16]
         | (1u << 16));                               // tensor_dim1 = 1
    g1[3] = (int)((nelem & 0xFFFFu) << 16);           // tile_dim0
    g1[4] = 0;                                        // tile_dim1/2 = 0 (1-D)
    g1[5] = (int)nelem;                               // tensor_dim0_stride lo
    g1[6] = 0;
    g1[7] = 0;
    const i32x4 z4 = {0, 0, 0, 0};
#if defined(__clang_major__) && (__clang_major__ >= 23)
    const i32x8 z8 = {0, 0, 0, 0, 0, 0, 0, 0};
    __builtin_amdgcn_tensor_load_to_lds(g0, g1, z4, z4, z8, 0);
#else
    __builtin_amdgcn_tensor_load_to_lds(g0, g1, z4, z4, 0);
#endif
#else
    (void)gsrc; (void)lds_off; (void)nelem;
#endif
}

__global__ __launch_bounds__(THREADS)
void survey_embeddings_kernel(const int*   __restrict__ year,      // [B]
                              const int*   __restrict__ answer,    // [B,Q]
                              const float* __restrict__ ans_emb,   // [VOCAB,D]
                              const float* __restrict__ lin_w,     // [D]
                              const float* __restrict__ lin_b,     // [D]
                              const float* __restrict__ ln_g,      // [D]
                              const float* __restrict__ ln_b,      // [D]
                              const float* __restrict__ year_emb,  // [NY,D]
                              const float* __restrict__ q_emb,     // [Q,D]
                              const float* __restrict__ alpha_p,   // [1]
                              const float* __restrict__ beta_p,    // [1]
                              float*       __restrict__ out,       // [B,Q,D]
                              int Q) {
    __shared__ float sh_q[QT * D_DIM];   // 16 KB: q_emb tile
    __shared__ float sh_y[BT * D_DIM];   //  1 KB: alpha * year_emb[year[b]]

    const int tid = threadIdx.x;
    const int q0  = blockIdx.x * QT;
    const int b0  = blockIdx.y * BT;

    const float alpha = alpha_p[0];
    const float beta  = beta_p[0];

    // ---- stage q_emb tile (contiguous 16 KB) into LDS ----
#if USE_TDM
    if (tid < 32) {   // one wave issues the DMA; TENSORcnt is per-wave
        tdm_load_1d_f32(q_emb + (size_t)q0 * D_DIM,
                        (unsigned)(uintptr_t)(void*)sh_q,
                        QT * D_DIM);
        wait_tensorcnt0();
    }
#else
    {
        const f32x4* src = (const f32x4*)(q_emb + (size_t)q0 * D_DIM);
        f32x4*       dst = (f32x4*)sh_q;
        for (int i = tid; i < (QT * D_DIM) / 4; i += THREADS) dst[i] = src[i];
    }
#endif

    // ---- precompute scaled yearly rows: 16 b's x 16 dims = 256 = THREADS ----
    {
        const int bi = tid >> 4;
        const int d  = tid & 15;
        sh_y[tid] = alpha * year_emb[(size_t)year[b0 + bi] * D_DIM + d];
    }
    __syncthreads();

    // ---- per-thread constant vectors (L2-resident broadcasts) ----
    f32x4 w[4], lb[4], g[4], nb[4];
#pragma unroll
    for (int k = 0; k < 4; ++k) {
        w[k]  = ((const f32x4*)lin_w)[k];
        lb[k] = ((const f32x4*)lin_b)[k];
        g[k]  = ((const f32x4*)ln_g)[k];
        nb[k] = ((const f32x4*)ln_b)[k];
    }

    const f32x4* qv = (const f32x4*)(sh_q + tid * D_DIM);

#pragma unroll 1
    for (int bi = 0; bi < BT; ++bi) {
        const int b = b0 + bi;
        const int q = q0 + tid;

        const int   ans  = answer[(size_t)b * Q + q];   // coalesced b32
        const bool  cont = (ans <= 1);
        const float a    = (float)ans;

        // categorical gather (64 B, L2-resident table); always load, select after
        const f32x4* ae = (const f32x4*)(ans_emb + (size_t)ans * D_DIM);
        f32x4 x[4];
#pragma unroll
        for (int k = 0; k < 4; ++k) {
            f32x4 cat = ae[k];
            f32x4 lin = a * w[k] + lb[k];
            x[k] = cont ? lin : cat;
        }

        // LayerNorm over D=16, all in registers
        float s = 0.f;
#pragma unroll
        for (int k = 0; k < 4; ++k)
#pragma unroll
            for (int j = 0; j < 4; ++j) s += x[k][j];
        const float mu = s * (1.0f / 16.0f);

        float ss = 0.f;
#pragma unroll
        for (int k = 0; k < 4; ++k)
#pragma unroll
            for (int j = 0; j < 4; ++j) {
                const float df = x[k][j] - mu;
                ss += df * df;
            }
        const float rs = rsqrtf(ss * (1.0f / 16.0f) + LN_EPS);

        const f32x4* yv = (const f32x4*)(sh_y + bi * D_DIM);  // broadcast DS read
        float* op = out + ((size_t)b * Q + q) * D_DIM;
#pragma unroll
        for (int k = 0; k < 4; ++k) {
            f32x4 o = (x[k] - mu) * rs * g[k] + nb[k] + yv[k] + beta * qv[k];
            __builtin_nontemporal_store(o, ((f32x4*)op) + k);  // streaming b128
        }
    }
}

extern "C" void kernel_launch(void* const* d_in, const int* in_sizes, int n_in,
                              void* d_out, int out_size, void* d_ws, size_t ws_size,
                              hipStream_t stream) {
    const int*   year     = (const int*)  d_in[0];
    const int*   answer   = (const int*)  d_in[1];
    const float* ans_emb  = (const float*)d_in[2];
    const float* lin_w    = (const float*)d_in[3];
    const float* lin_b    = (const float*)d_in[4];
    const float* ln_g     = (const float*)d_in[5];
    const float* ln_b     = (const float*)d_in[6];
    const float* year_emb = (const float*)d_in[7];
    const float* q_emb    = (const float*)d_in[8];
    const float* alpha    = (const float*)d_in[9];
    const float* beta     = (const float*)d_in[10];
    float*       out      = (float*)d_out;

    const int B = in_sizes[0];
    const int Q = in_sizes[1] / B;

    dim3 grid((unsigned)(Q / QT), (unsigned)(B / BT));
    survey_embeddings_kernel<<<grid, dim3(THREADS), 0, stream>>>(
        year, answer, ans_emb, lin_w, lin_b, ln_g, ln_b,
        year_emb, q_emb, alpha, beta, out, Q);

    (void)n_in; (void)out_size; (void)d_ws; (void)ws_size;
}